// MultiTaskLossWrapper_14121852470100
// MI455X (gfx1250) — compile-verified
//
#include <hip/hip_runtime.h>
#include <hip/hip_bf16.h>

typedef __attribute__((ext_vector_type(2))) float v2f;
typedef __attribute__((ext_vector_type(8))) float v8f;

// ---------------------------------------------------------------------------
// Full 32-lane sum via v_wmma_f32_16x16x4_f32 with an all-ones B matrix.
// A layout (f32 16x4): lanes 0-15 hold (M=lane, K=0/1) in VGPR0/1,
// lanes 16-31 hold (M=lane-16, K=2/3). With A = {x, 0} and B = ones:
//   D[m][n] = x_m + x_{m+16}  (independent of n)
// Lane L<16 holds D[0..7][L] in its 8 C VGPRs, lane L>=16 holds D[8..15][L-16].
// So sum(c[0..7]) + shfl_down(.,16) = sum over all 32 lanes.
// Requires EXEC == all ones (call only from reconverged, full-wave code).
// ---------------------------------------------------------------------------
__device__ __forceinline__ float wave_sum_wmma(float x) {
    v2f a; a.x = x;    a.y = 0.0f;
    v2f b; b.x = 1.0f; b.y = 1.0f;
    v8f c = {};
    c = __builtin_amdgcn_wmma_f32_16x16x4_f32(
        /*neg_a=*/false, a, /*neg_b=*/false, b,
        /*c_mod=*/(short)0, c, /*reuse_a=*/false, /*reuse_b=*/false);
    float s = c[0] + c[1] + c[2] + c[3] + c[4] + c[5] + c[6] + c[7];
    s += __shfl_down(s, 16);
    return s;  // valid in lanes 0..15
}

// Numerically stable softplus with fast transcendentals.
__device__ __forceinline__ float softplus_neg_abs_based(float x) {
    // softplus(x) = max(x,0) + log(1 + exp(-|x|))
    float m = fmaxf(x, 0.0f);
    return m + __logf(1.0f + __expf(-fabsf(x)));
}

__device__ __forceinline__ void process_ann(float p0, float p1, int lab, float w,
                                            float& acc, float& cnt) {
    // t = clip(lab, 0, 1) with lab in {-1,0,1}  ->  t = (lab == 1)
    float t  = (lab == 1) ? 1.0f : 0.0f;
    float u  = 1.0f - t;
    // softplus(-p) computed once; softplus(p) = p + softplus(-p)
    float sm0 = softplus_neg_abs_based(-p0);
    float sp0 = sm0 + p0;
    float sm1 = softplus_neg_abs_based(-p1);
    float sp1 = sm1 + p1;
    // elem = w * target * softplus(-pred) + (1 - target) * softplus(pred)
    float e0 = w * u * sm0 + t * sp0;   // target0 = 1-t
    float e1 = w * t * sm1 + u * sp1;   // target1 = t
    float per_ann = 0.5f * (e0 + e1);
    float m = (lab != -1) ? 1.0f : 0.0f;
    acc += m * per_ann;
    cnt += m;
}

// ---------------------------------------------------------------------------
// Kernel 1: one wave32 per sample row b.
//   sums[b]  = has_valid ? sum_a mask * per_ann : 0
//   flags[b] = has_valid ? 1 : 0
// ---------------------------------------------------------------------------
#define WAVES_PER_BLOCK 8
#define MAX_A_LDS 512

__global__ void mtl_row_kernel(const float* __restrict__ preds,
                               const int*   __restrict__ labels,
                               const float* __restrict__ weights,
                               float* __restrict__ sums,
                               float* __restrict__ flags,
                               int B, int A) {
    __shared__ float sw[MAX_A_LDS];

    const int tid  = threadIdx.x;
    const int lane = tid & 31;
    const int wid  = tid >> 5;

    // Stage weights into LDS (all waves cooperate) when they fit.
    const bool use_lds = (A <= MAX_A_LDS);
    if (use_lds) {
        for (int t = tid; t < A; t += blockDim.x) sw[t] = weights[t];
    }
    __syncthreads();

    const int b = blockIdx.x * WAVES_PER_BLOCK + wid;
    if (b >= B) return;  // wave-uniform exit (after the barrier)

    const float* prow = preds  + (size_t)b * (size_t)A * 2u;
    const int*   lrow = labels + (size_t)b * (size_t)A;

    float acc = 0.0f;
    float cnt = 0.0f;

    // 2 annotations per lane per iteration: float4 preds + int2 labels,
    // both coalesced and naturally aligned (a0 even, A even).
    int a0 = lane * 2;
    for (; a0 + 1 < A; a0 += 64) {
        // Prefetch the next strip of this row's stream (speculative, CDNA5).
        __builtin_prefetch(prow + 2 * a0 + 128, 0, 0);
        __builtin_prefetch(lrow + a0 + 64, 0, 0);

        float4 p  = *reinterpret_cast<const float4*>(prow + 2 * a0);
        int2   lb = *reinterpret_cast<const int2*>(lrow + a0);
        float  w0 = use_lds ? sw[a0]     : weights[a0];
        float  w1 = use_lds ? sw[a0 + 1] : weights[a0 + 1];
        process_ann(p.x, p.y, lb.x, w0, acc, cnt);
        process_ann(p.z, p.w, lb.y, w1, acc, cnt);
    }
    if (a0 < A) {  // odd-A tail (not taken for A=512)
        float2 p  = *reinterpret_cast<const float2*>(prow + 2 * a0);
        int    lb = lrow[a0];
        float  w0 = use_lds ? sw[a0] : weights[a0];
        process_ann(p.x, p.y, lb, w0, acc, cnt);
    }

    // Reconverged full wave here: WMMA cross-lane reductions.
    float row_sum = wave_sum_wmma(acc);
    float row_cnt = wave_sum_wmma(cnt);

    if (lane == 0) {
        float valid = (row_cnt > 0.0f) ? 1.0f : 0.0f;
        sums[b]  = valid * row_sum;
        flags[b] = valid;
    }
}

// ---------------------------------------------------------------------------
// Kernel 2: single-wave deterministic final reduction.
//   out = sum(sums) / max(sum(flags), 1)
// ---------------------------------------------------------------------------
__global__ void mtl_final_kernel(const float* __restrict__ sums,
                                 const float* __restrict__ flags,
                                 float* __restrict__ out, int n) {
    const int lane = threadIdx.x;
    float s = 0.0f, c = 0.0f;
    for (int i = lane; i < n; i += 32) {
        s += sums[i];
        c += flags[i];
    }
    float total_sum = wave_sum_wmma(s);
    float total_cnt = wave_sum_wmma(c);
    if (lane == 0) {
        float n_valid = fmaxf(total_cnt, 1.0f);
        out[0] = total_sum / n_valid;
    }
}

extern "C" void kernel_launch(void* const* d_in, const int* in_sizes, int n_in,
                              void* d_out, int out_size, void* d_ws, size_t ws_size,
                              hipStream_t stream) {
    const float* preds   = (const float*)d_in[0];   // (B, A, 2) f32
    const int*   labels  = (const int*)d_in[1];     // (B, A) int32 (JAX x64 off)
    const float* weights = (const float*)d_in[2];   // (A,) f32

    const int A = in_sizes[2];
    const int B = (A > 0) ? (in_sizes[1] / A) : 0;

    float* sums  = (float*)d_ws;      // B floats
    float* flags = sums + B;          // B floats

    const int blocks = (B + WAVES_PER_BLOCK - 1) / WAVES_PER_BLOCK;
    mtl_row_kernel<<<blocks, WAVES_PER_BLOCK * 32, 0, stream>>>(
        preds, labels, weights, sums, flags, B, A);
    mtl_final_kernel<<<1, 32, 0, stream>>>(sums, flags, (float*)d_out, B);
}